// SpikeFP64Floor_29875792511520
// MI455X (gfx1250) — compile-verified
//
#include <hip/hip_runtime.h>

// Row = 64 float32 "bits" (MSB first) of one float64. One thread per row.
// 256 B in + 256 B out per row -> pure HBM-streaming kernel (~22 us roofline
// at 23.3 TB/s for 512 MB total).
//
// CDNA5 path: stage each row into LDS with the async engine
// (GLOBAL_LOAD_ASYNC_TO_LDS_B128 th:TH_LOAD_NT, ASYNCcnt-tracked, no VGPR
// return data), consume via ds_load_b128, write results with non-temporal
// B128 stores. Both 256 MB streams are touched exactly once, so NT policy
// keeps them from thrashing the 192 MB global L2.

typedef float f4 __attribute__((ext_vector_type(4)));

__global__ __launch_bounds__(256) void floor_bits_kernel(
    const float* __restrict__ x, float* __restrict__ out, int nrows) {
  // 256 threads * 256 B = 64 KB LDS tile (one row per thread).
  __shared__ float tile[256 * 64];

  const int row = blockIdx.x * blockDim.x + threadIdx.x;
  if (row >= nrows) return;

  float* lrow = tile + (size_t)threadIdx.x * 64;
  // Low 32 bits of a generic LDS pointer = byte offset within the
  // workgroup's LDS allocation (aperture bits live in the high half).
  const unsigned ldsa = (unsigned)(uintptr_t)lrow;
  const unsigned gvo = (unsigned)row * 256u;  // byte offset from x (fits 32b)

  // GVS mode: mem = SADDR + VADDR + INST_OFFSET; lds = VDST + INST_OFFSET.
  // 16 x B128 covers the 256-byte row; one wait drains ASYNCcnt.
  asm volatile(
      "global_load_async_to_lds_b128 %0, %1, %2 offset:0 th:TH_LOAD_NT\n\t"
      "global_load_async_to_lds_b128 %0, %1, %2 offset:16 th:TH_LOAD_NT\n\t"
      "global_load_async_to_lds_b128 %0, %1, %2 offset:32 th:TH_LOAD_NT\n\t"
      "global_load_async_to_lds_b128 %0, %1, %2 offset:48 th:TH_LOAD_NT\n\t"
      "global_load_async_to_lds_b128 %0, %1, %2 offset:64 th:TH_LOAD_NT\n\t"
      "global_load_async_to_lds_b128 %0, %1, %2 offset:80 th:TH_LOAD_NT\n\t"
      "global_load_async_to_lds_b128 %0, %1, %2 offset:96 th:TH_LOAD_NT\n\t"
      "global_load_async_to_lds_b128 %0, %1, %2 offset:112 th:TH_LOAD_NT\n\t"
      "global_load_async_to_lds_b128 %0, %1, %2 offset:128 th:TH_LOAD_NT\n\t"
      "global_load_async_to_lds_b128 %0, %1, %2 offset:144 th:TH_LOAD_NT\n\t"
      "global_load_async_to_lds_b128 %0, %1, %2 offset:160 th:TH_LOAD_NT\n\t"
      "global_load_async_to_lds_b128 %0, %1, %2 offset:176 th:TH_LOAD_NT\n\t"
      "global_load_async_to_lds_b128 %0, %1, %2 offset:192 th:TH_LOAD_NT\n\t"
      "global_load_async_to_lds_b128 %0, %1, %2 offset:208 th:TH_LOAD_NT\n\t"
      "global_load_async_to_lds_b128 %0, %1, %2 offset:224 th:TH_LOAD_NT\n\t"
      "global_load_async_to_lds_b128 %0, %1, %2 offset:240 th:TH_LOAD_NT\n\t"
      "s_wait_asynccnt 0x0"
      :
      : "v"(ldsa), "v"(gvo), "s"(x)
      : "memory");

  // Consume the staged row from LDS (ds_load_b128).
  const f4* __restrict__ lr = reinterpret_cast<const f4*>(lrow);
  f4 v[16];
#pragma unroll
  for (int i = 0; i < 16; ++i) {
    v[i] = lr[i];
  }

  // Pack 64 bits, x[k] -> bit (63-k).
  unsigned long long u = 0ull;
#pragma unroll
  for (int i = 0; i < 16; ++i) {
    const int b = 63 - 4 * i;
    u |= (v[i].x > 0.5f) ? (1ull << b) : 0ull;
    u |= (v[i].y > 0.5f) ? (1ull << (b - 1)) : 0ull;
    u |= (v[i].z > 0.5f) ? (1ull << (b - 2)) : 0ull;
    u |= (v[i].w > 0.5f) ? (1ull << (b - 3)) : 0ull;
  }

  // floor() on the raw float64 bit pattern (bit-exact replica of reference).
  const unsigned long long sign = u & 0x8000000000000000ull;
  const int e = (int)((u >> 52) & 0x7ffu);
  unsigned long long r;
  if (e >= 1075) {
    // Already integral (or inf/NaN): unchanged.
    r = u;
  } else if (e < 1023) {
    // |x| < 1: reference maps sign=1 -> -1.0 bits, sign=0 -> +0.0.
    r = sign ? 0xBFF0000000000000ull : 0x0ull;
  } else {
    const unsigned long long fmask = 0x000FFFFFFFFFFFFFull >> (e - 1023);
    const unsigned long long tu = u & ~fmask;  // truncate toward zero
    if (sign && (u & fmask)) {
      // Negative non-integer: floor = trunc - 1.0 (exact, |trunc| < 2^52).
      r = (unsigned long long)__double_as_longlong(
          __longlong_as_double((long long)tu) - 1.0);
    } else {
      r = tu;
    }
  }

  // Unpack bit (63-k) -> out[k] as 0.0f / 1.0f, NT B128 stores.
  const unsigned int rhi = (unsigned int)(r >> 32);
  const unsigned int rlo = (unsigned int)r;
  f4* __restrict__ orow = reinterpret_cast<f4*>(out + (size_t)row * 64);
#pragma unroll
  for (int i = 0; i < 16; ++i) {
    const int b = 63 - 4 * i;  // bit index of component .x
    f4 o;
    if (b >= 35) {  // all four bits come from rhi (b-3 >= 32)
      o.x = ((rhi >> (b - 32)) & 1u) ? 1.0f : 0.0f;
      o.y = ((rhi >> (b - 33)) & 1u) ? 1.0f : 0.0f;
      o.z = ((rhi >> (b - 34)) & 1u) ? 1.0f : 0.0f;
      o.w = ((rhi >> (b - 35)) & 1u) ? 1.0f : 0.0f;
    } else {  // b <= 31: all four bits from rlo
      o.x = ((rlo >> b) & 1u) ? 1.0f : 0.0f;
      o.y = ((rlo >> (b - 1)) & 1u) ? 1.0f : 0.0f;
      o.z = ((rlo >> (b - 2)) & 1u) ? 1.0f : 0.0f;
      o.w = ((rlo >> (b - 3)) & 1u) ? 1.0f : 0.0f;
    }
    __builtin_nontemporal_store(o, orow + i);
  }
}

extern "C" void kernel_launch(void* const* d_in, const int* in_sizes, int n_in,
                              void* d_out, int out_size, void* d_ws,
                              size_t ws_size, hipStream_t stream) {
  const float* x = (const float*)d_in[0];
  float* out = (float*)d_out;
  const int nrows = in_sizes[0] / 64;  // 1048576
  const int threads = 256;             // 8 wave32 per block
  const int blocks = (nrows + threads - 1) / threads;
  floor_bits_kernel<<<blocks, threads, 0, stream>>>(x, out, nrows);
}